// AtIndexPooler_24515673325653
// MI455X (gfx1250) — compile-verified
//
#include <hip/hip_runtime.h>

// AtIndexPooler: out[b, j, :] = (indices[b,j] >= 0)
//                  ? hidden_state[b, clip(indices[b,j]+OFFSET, 0, SEQ-1), :]
//                  : missing_embeddings[j, :]
// Shapes: hidden_state (512, 512, 1024) f32, missing (2, 1024) f32,
//         indices (512, 2) int, out (512, 2*1024) f32.
//
// Pure gather: ~8 MB of traffic, 0 FLOPs. One workgroup per (batch, index)
// pair; 256 lanes x float4 = one 4 KB row per block, b128 loads/stores with
// non-temporal hints (rows are touched exactly once). Index load is uniform
// per block -> scalar SMEM load.

constexpr int BATCH       = 512;
constexpr int SEQ_LEN     = 512;
constexpr int HIDDEN      = 1024;
constexpr int NUM_INDICES = 2;
constexpr int OFFSET      = 0;

typedef __attribute__((ext_vector_type(4))) float v4f;

__global__ __launch_bounds__(256) void
AtIndexPooler_gather_kernel(const float* __restrict__ hidden,
                            const float* __restrict__ missing,
                            const int*   __restrict__ indices,
                            float*       __restrict__ out)
{
    const int bj = blockIdx.x;              // 0 .. BATCH*NUM_INDICES-1
    const int b  = bj / NUM_INDICES;
    const int j  = bj - b * NUM_INDICES;

    // Uniform across the block -> compiler emits s_load_b32 (SMEM path).
    const int  raw   = indices[bj];         // indices[b*NUM_INDICES + j]
    const bool valid = (raw >= 0);

    int idx = raw + OFFSET;
    idx = idx < 0 ? 0 : idx;
    idx = idx > (SEQ_LEN - 1) ? (SEQ_LEN - 1) : idx;

    const float* src = valid
        ? hidden  + ((size_t)b * SEQ_LEN + (size_t)idx) * HIDDEN
        : missing + (size_t)j * HIDDEN;
    float* dst = out + (size_t)bj * HIDDEN;

    // 256 lanes * 4 floats = 1024 = HIDDEN: one b128 load + one b128 store
    // per lane, fully coalesced, non-temporal (single-touch data).
    const int t = threadIdx.x;
    v4f v = __builtin_nontemporal_load(reinterpret_cast<const v4f*>(src) + t);
    __builtin_nontemporal_store(v, reinterpret_cast<v4f*>(dst) + t);
}

extern "C" void kernel_launch(void* const* d_in, const int* in_sizes, int n_in,
                              void* d_out, int out_size, void* d_ws, size_t ws_size,
                              hipStream_t stream)
{
    (void)in_sizes; (void)n_in; (void)out_size; (void)d_ws; (void)ws_size;

    const float* hidden  = (const float*)d_in[0];  // (512, 512, 1024) f32
    const float* missing = (const float*)d_in[1];  // (2, 1024) f32
    const int*   indices = (const int*)d_in[2];    // (512, 2) int
    float*       out     = (float*)d_out;          // (512, 2048) f32

    const dim3 grid(BATCH * NUM_INDICES);          // 1024 blocks
    const dim3 block(HIDDEN / 4);                  // 256 threads = 8 waves (wave32)

    AtIndexPooler_gather_kernel<<<grid, block, 0, stream>>>(hidden, missing, indices, out);
}